// CrossEntropyLoss_55379308315499
// MI455X (gfx1250) — compile-verified
//
#include <hip/hip_runtime.h>
#include <cstdint>
#include <cstddef>

// CrossEntropyLoss over [B=8192, V=32000] f32 logits, int64 labels.
// loss = ( -sum_i out[i,lab[i]] + sum_i log(sum_j exp(out[i,j]) + 1e-5) ) / B
//
// Memory-bound: ~1.05 GB @ 23.3 TB/s => ~45us floor. Strategy: stream each row
// through LDS with CDNA5 async global->LDS loads (ASYNCcnt) in a double
// buffer, exp on the TRANS pipe, wave32 shuffle reduction.

#define THREADS 256
#define WAVESZ  32
#define NWAVES  (THREADS / WAVESZ)
#define CHUNK_FLOATS (THREADS * 4)   // 1024 floats = 4KB per chunk
#define CHUNK_BYTES  (CHUNK_FLOATS * 4)

// ---- CDNA5 async copy: 16 bytes/lane, global -> LDS, tracked by ASYNCcnt ----
__device__ __forceinline__ void async_copy_b128(uint32_t lds_byte_addr,
                                                const void* global_addr) {
  // VDST VGPR = LDS byte address, VADDR = 64-bit global VA (GV mode).
  asm volatile("global_load_async_to_lds_b128 %0, %1, off"
               :
               : "v"(lds_byte_addr), "v"(global_addr)
               : "memory");
}

__device__ __forceinline__ void wait_asynccnt_le1() {
  asm volatile("s_wait_asynccnt 1" ::: "memory");
}
__device__ __forceinline__ void wait_asynccnt_0() {
  asm volatile("s_wait_asynccnt 0" ::: "memory");
}

__device__ __forceinline__ float wave_reduce_add(float v) {
  #pragma unroll
  for (int off = WAVESZ / 2; off > 0; off >>= 1)
    v += __shfl_xor(v, off, WAVESZ);
  return v;
}

// One block per row: rowloss[row] = log(sum_j exp(x[row,j]) + 1e-5) - x[row,lab[row]]
__global__ __launch_bounds__(THREADS)
void ce_row_kernel(const float* __restrict__ logits,
                   const long long* __restrict__ label,
                   float* __restrict__ rowloss, int V) {
  extern __shared__ float smem[];            // [2*1024] stage | [NWAVES] reduce
  const int row = blockIdx.x;
  const int tid = threadIdx.x;
  const float* rowp = logits + (size_t)row * (size_t)V;

  // Low 32 bits of a generic (flat) LDS pointer are the LDS byte address
  // (aperture rule, ISA 10.2), usable directly as VDST for the async load.
  const uint32_t lds_base = (uint32_t)(uintptr_t)(void*)smem;

  const int NC = V / CHUNK_FLOATS;           // full async chunks (31 for V=32000)
  float acc = 0.0f;

  if (NC > 0) {
    // Prologue: chunk 0 into buffer 0.
    async_copy_b128(lds_base + (uint32_t)(tid * 16), rowp + (size_t)tid * 4);

    for (int c = 0; c < NC; ++c) {
      const int cur = c & 1;
      if (c + 1 < NC) {
        const int nb = (c + 1) & 1;
        async_copy_b128(lds_base + (uint32_t)(nb * CHUNK_BYTES + tid * 16),
                        rowp + (size_t)(c + 1) * CHUNK_FLOATS + (size_t)tid * 4);
        wait_asynccnt_le1();                 // in-order: chunk c has landed
      } else {
        wait_asynccnt_0();
      }
      __syncthreads();                       // chunk c visible to all waves

      const float4 v =
          *reinterpret_cast<const float4*>(&smem[cur * CHUNK_FLOATS + tid * 4]);
      acc += __expf(v.x);                    // v_exp_f32 on TRANS pipe,
      acc += __expf(v.y);                    // v_add_f32 co-executes on VALU
      acc += __expf(v.z);
      acc += __expf(v.w);

      __syncthreads();                       // done reading before overwrite
    }
  }
  // Tail (256 floats for V=32000): plain global loads.
  for (int j = NC * CHUNK_FLOATS + tid; j < V; j += THREADS)
    acc += __expf(rowp[j]);

  // Block reduction: wave32 shuffle, then across the 8 waves via LDS.
  acc = wave_reduce_add(acc);
  float* red = smem + 2 * CHUNK_FLOATS;
  const int lane = tid & (WAVESZ - 1);
  const int wv   = tid >> 5;
  if (lane == 0) red[wv] = acc;
  __syncthreads();
  if (tid == 0) {
    float tot = 0.0f;
    #pragma unroll
    for (int w = 0; w < NWAVES; ++w) tot += red[w];
    const long long l = label[row];          // int64 labels per reference
    const float gathered = rowp[l];
    rowloss[row] = __logf(tot + 1e-5f) - gathered;
  }
}

// Deterministic single-block final reduction: d_out[0] = sum(rowloss)/B.
__global__ __launch_bounds__(THREADS)
void ce_final_kernel(const float* __restrict__ rowloss,
                     float* __restrict__ out, int B) {
  __shared__ float red[NWAVES];
  float s = 0.0f;
  for (int i = threadIdx.x; i < B; i += THREADS) s += rowloss[i];
  s = wave_reduce_add(s);
  const int lane = threadIdx.x & (WAVESZ - 1);
  const int wv   = threadIdx.x >> 5;
  if (lane == 0) red[wv] = s;
  __syncthreads();
  if (threadIdx.x == 0) {
    float tot = 0.0f;
    #pragma unroll
    for (int w = 0; w < NWAVES; ++w) tot += red[w];
    out[0] = tot / (float)B;
  }
}

extern "C" void kernel_launch(void* const* d_in, const int* in_sizes, int n_in,
                              void* d_out, int out_size, void* d_ws, size_t ws_size,
                              hipStream_t stream) {
  const float*     logits = (const float*)d_in[0];
  const long long* label  = (const long long*)d_in[1];   // jnp.int64
  const int B = in_sizes[1];                              // 8192
  const int V = in_sizes[0] / B;                          // 32000
  float* rowloss = (float*)d_ws;                          // B floats (rewritten every call)

  const size_t shmem = (size_t)(2 * CHUNK_FLOATS + NWAVES) * sizeof(float); // 8.2KB
  ce_row_kernel<<<B, THREADS, shmem, stream>>>(logits, label, rowloss, V);
  ce_final_kernel<<<1, THREADS, 0, stream>>>(rowloss, (float*)d_out, B);
}